// KANPERCE_87213605913229
// MI455X (gfx1250) — compile-verified
//
#include <hip/hip_runtime.h>
#include <hip/hip_bf16.h>
#include <math.h>

// ---------------------------------------------------------------------------
// KAN-perceptron forward for MI455X (gfx1250, wave32).
//   h   = x @ W1^T + b1                         (fp32 WMMA GEMM)
//   L1  : F1 = [silu(h_i), Bspline(h_i,grid1)]  -> out1 = F1 @ W1L (WMMA GEMM)
//   L2  : F2 = [silu(o_i), Bspline(o_i,grid2)]  -> out  = F2 @ W2L (WMMA GEMM)
// Matrix path: V_WMMA_F32_16X16X4_F32 (fp32, matches reference precision).
// Data path: GLOBAL_LOAD_ASYNC_TO_LDS_B128 double-buffered pipeline (ASYNCcnt).
// ---------------------------------------------------------------------------

typedef __attribute__((ext_vector_type(2))) float v2f;
typedef __attribute__((ext_vector_type(4))) float v4f;
typedef __attribute__((ext_vector_type(8))) float v8f;
typedef __attribute__((ext_vector_type(4))) int   v4i;

#define BDIM     4096
#define INDIM    1024
#define HID      256
#define OUTDIM   128
#define KC       8      // NUM + K spline coefficients per edge
#define GPTS     12     // grid points per channel (NUM+1 + 2K)
#define FEAT     9      // 1 (silu base) + KC spline bases
#define F1K      (HID * FEAT)   // 2304, K-dim of the layer GEMMs

// ---- gfx1250 async global->LDS copy (ASYNCcnt path), with safe fallbacks ---
#if defined(__has_builtin)
#  if __has_builtin(__builtin_amdgcn_global_load_async_to_lds_b128)
#    define HAVE_ASYNC_LDS 1
#  endif
#endif

__device__ __forceinline__ void copy16_g2lds(const float* g, float* lds)
{
#if defined(HAVE_ASYNC_LDS)
    __builtin_amdgcn_global_load_async_to_lds_b128(
        (__attribute__((address_space(1))) v4i*)g,
        (__attribute__((address_space(3))) v4i*)lds,
        0, 0);
#else
    *(v4f*)lds = *(const v4f*)g;   // sync fallback through VGPRs
#endif
}

__device__ __forceinline__ void async_drain()
{
#if defined(HAVE_ASYNC_LDS)
#  if __has_builtin(__builtin_amdgcn_s_wait_asynccnt)
    __builtin_amdgcn_s_wait_asynccnt(0);
#  else
    asm volatile("s_wait_asynccnt 0x0" ::: "memory");
#  endif
#endif
}

// ---------------------------------------------------------------------------
// fp32 WMMA GEMM:  C[M,N] = A[M,K] (row major) * B[K,N] (row major) + bias[N]
// M multiple of 128, N multiple of 64, K multiple of 16.
// Block: 256 threads (8 waves), 128x64 tile; wave -> 32x32 (four 16x16 accs).
// Double-buffered LDS, async global->LDS copies overlapped with WMMA.
// ---------------------------------------------------------------------------
#define BLK_M  128
#define BLK_N  64
#define TILE_K 16
#define LDA_PAD 20   // floats; row stride 80B: 16B-aligned, spreads LDS banks

__global__ __launch_bounds__(256)
void gemm_f32_wmma(const float* __restrict__ A, const float* __restrict__ B,
                   const float* __restrict__ bias, float* __restrict__ C,
                   int M, int N, int K)
{
    __shared__ float sA[2][BLK_M * LDA_PAD];
    __shared__ float sB[2][TILE_K * BLK_N];

    const int tid   = threadIdx.x;
    const int lane  = tid & 31;
    const int wave  = tid >> 5;
    const int waveM = wave & 3;        // 4 waves along M
    const int waveN = wave >> 2;       // 2 waves along N
    const int bM    = blockIdx.y * BLK_M;
    const int bN    = blockIdx.x * BLK_N;
    const int lrow  = lane & 15;       // lane row/col within a 16-tile
    const int khalf = lane >> 4;       // 0: K{0,1}/M{0..7}, 1: K{2,3}/M{8..15}

    v8f acc00 = {}, acc01 = {}, acc10 = {}, acc11 = {};

    // cooperative tile-load mapping (one 16B chunk per thread per sub-tile)
    const int arow = tid >> 2, ac4 = (tid & 3) << 2;   // A: 2x(64 rows x 16 k)
    const int brow = tid >> 4, bc4 = (tid & 15) << 2;  // B: 16 k x 64 cols
    const float* Ard0 = A + (size_t)(bM + arow)      * K + ac4;
    const float* Ard1 = A + (size_t)(bM + arow + 64) * K + ac4;
    const float* Brd  = B + (size_t)brow * N + bN + bc4;

    // prologue: stage tile 0 into buffer 0
    copy16_g2lds(Ard0, &sA[0][arow * LDA_PAD + ac4]);
    copy16_g2lds(Ard1, &sA[0][(arow + 64) * LDA_PAD + ac4]);
    copy16_g2lds(Brd,  &sB[0][brow * BLK_N + bc4]);
    async_drain();
    __syncthreads();

    int buf = 0;
    for (int k0 = 0; k0 < K; k0 += TILE_K) {
        // issue async copies for the NEXT slice into the other buffer
        if (k0 + TILE_K < K) {
            const int kn = k0 + TILE_K;
            copy16_g2lds(Ard0 + kn, &sA[buf ^ 1][arow * LDA_PAD + ac4]);
            copy16_g2lds(Ard1 + kn, &sA[buf ^ 1][(arow + 64) * LDA_PAD + ac4]);
            copy16_g2lds(Brd + (size_t)kn * N, &sB[buf ^ 1][brow * BLK_N + bc4]);
        }

        // compute on current buffer: 4 k-steps x 4 WMMAs
        const float* cA = sA[buf];
        const float* cB = sB[buf];
        #pragma unroll
        for (int s = 0; s < TILE_K / 4; ++s) {
            const int kk = s * 4 + khalf * 2;          // this lane's K pair
            v2f a0 = *(const v2f*)&cA[(waveM * 32 +      lrow) * LDA_PAD + kk];
            v2f a1 = *(const v2f*)&cA[(waveM * 32 + 16 + lrow) * LDA_PAD + kk];
            const int c0 = waveN * 32 + lrow;
            v2f b0, b1;
            b0.x = cB[kk * BLK_N + c0];           b0.y = cB[(kk + 1) * BLK_N + c0];
            b1.x = cB[kk * BLK_N + c0 + 16];      b1.y = cB[(kk + 1) * BLK_N + c0 + 16];
            acc00 = __builtin_amdgcn_wmma_f32_16x16x4_f32(false, a0, false, b0, (short)0, acc00, false, false);
            acc01 = __builtin_amdgcn_wmma_f32_16x16x4_f32(false, a0, false, b1, (short)0, acc01, false, false);
            acc10 = __builtin_amdgcn_wmma_f32_16x16x4_f32(false, a1, false, b0, (short)0, acc10, false, false);
            acc11 = __builtin_amdgcn_wmma_f32_16x16x4_f32(false, a1, false, b1, (short)0, acc11, false, false);
        }

        async_drain();       // own async writes to next buffer are in LDS
        __syncthreads();     // everyone done reading current / writing next
        buf ^= 1;
    }

    // C/D layout: lane<16 -> VGPR r = C[M=r][N=lane]; lane>=16 -> M=r+8
    const int gc0 = bN + waveN * 32 + lrow;
    const int gc1 = gc0 + 16;
    const float bv0 = bias ? bias[gc0] : 0.0f;
    const float bv1 = bias ? bias[gc1] : 0.0f;
    const int r0 = bM + waveM * 32 + khalf * 8;
    #pragma unroll
    for (int r = 0; r < 8; ++r) {
        C[(size_t)(r0 + r)      * N + gc0] = acc00[r] + bv0;
        C[(size_t)(r0 + r)      * N + gc1] = acc01[r] + bv1;
        C[(size_t)(r0 + 16 + r) * N + gc0] = acc10[r] + bv0;
        C[(size_t)(r0 + 16 + r) * N + gc1] = acc11[r] + bv1;
    }
}

// ---------------------------------------------------------------------------
// W1 transpose: W1 (HID x IN) row-major -> W1T (IN x HID) so the GEMM's
// B operand is K-major (lane-coalesced loads).
// ---------------------------------------------------------------------------
__global__ void transpose_w1(const float* __restrict__ W1, float* __restrict__ W1T)
{
    int idx = blockIdx.x * blockDim.x + threadIdx.x;   // k*HID + n
    if (idx >= INDIM * HID) return;
    int k = idx / HID, n = idx % HID;
    W1T[idx] = W1[(size_t)n * INDIM + k];
}

// ---------------------------------------------------------------------------
// Pack KAN layer weights:  W[(i*9 + 0), o] = mask*sb
//                          W[(i*9+1+c), o] = mask*sp*coef[i,o,c]
// ---------------------------------------------------------------------------
__global__ void pack_kan_w(const float* __restrict__ coef, const float* __restrict__ sb,
                           const float* __restrict__ sp,   const float* __restrict__ mask,
                           float* __restrict__ W, int I, int O)
{
    int idx = blockIdx.x * blockDim.x + threadIdx.x;   // i*O + o
    if (idx >= I * O) return;
    int i = idx / O, o = idx % O;
    float m   = mask[idx];
    float msp = m * sp[idx];
    W[(size_t)(i * FEAT) * O + o] = m * sb[idx];
    #pragma unroll
    for (int c = 0; c < KC; ++c)
        W[(size_t)(i * FEAT + 1 + c) * O + o] = msp * coef[(size_t)idx * KC + c];
}

// ---------------------------------------------------------------------------
// Featurize: F[b, i*9+0] = silu(h); F[b, i*9+1+c] = Bspline_c(h, grid_i)
// Cox–de Boor, k=3 over 12 knots -> 8 bases (uniform grid: no 0/0).
// ---------------------------------------------------------------------------
__global__ void kan_featurize(const float* __restrict__ H, const float* __restrict__ grid,
                              float* __restrict__ F, int Bn, int I)
{
    int idx = blockIdx.x * blockDim.x + threadIdx.x;   // b*I + i
    if (idx >= Bn * I) return;
    int i = idx % I;
    float x = H[idx];

    float g[GPTS];
    #pragma unroll
    for (int j = 0; j < GPTS; ++j) g[j] = grid[(size_t)i * GPTS + j];

    float Bv[GPTS - 1];
    #pragma unroll
    for (int j = 0; j < GPTS - 1; ++j)
        Bv[j] = (x >= g[j] && x < g[j + 1]) ? 1.0f : 0.0f;

    #pragma unroll
    for (int ki = 1; ki <= 3; ++ki) {
        #pragma unroll
        for (int j = 0; j < GPTS - 1 - ki; ++j) {
            float t1 = (x - g[j]) / (g[j + ki] - g[j]);
            float t2 = (g[j + ki + 1] - x) / (g[j + ki + 1] - g[j + 1]);
            Bv[j] = t1 * Bv[j] + t2 * Bv[j + 1];
        }
    }

    float sil = x * (1.0f / (1.0f + __expf(-x)));
    float* o = F + (size_t)(idx / I) * (I * FEAT) + (size_t)i * FEAT;
    o[0] = sil;
    #pragma unroll
    for (int c = 0; c < KC; ++c) o[1 + c] = Bv[c];
}

// ---------------------------------------------------------------------------
// Host-side orchestration (graph-capture safe: launches only).
// ---------------------------------------------------------------------------
extern "C" void kernel_launch(void* const* d_in, const int* in_sizes, int n_in,
                              void* d_out, int out_size, void* d_ws, size_t ws_size,
                              hipStream_t stream)
{
    const float* x     = (const float*)d_in[0];
    const float* W1    = (const float*)d_in[1];
    const float* b1    = (const float*)d_in[2];
    const float* grid1 = (const float*)d_in[3];
    const float* coef1 = (const float*)d_in[4];
    const float* sb1   = (const float*)d_in[5];
    const float* sp1   = (const float*)d_in[6];
    const float* mask1 = (const float*)d_in[7];
    const float* grid2 = (const float*)d_in[8];
    const float* coef2 = (const float*)d_in[9];
    const float* sb2   = (const float*)d_in[10];
    const float* sp2   = (const float*)d_in[11];
    const float* mask2 = (const float*)d_in[12];
    float* out = (float*)d_out;

    // workspace layout (floats), ~50.7 MB total
    float* ws   = (float*)d_ws;
    float* W1T  = ws;                          // 1024*256
    float* h    = W1T  + (size_t)INDIM * HID;  // 4096*256
    float* F    = h    + (size_t)BDIM * HID;   // 4096*2304 (reused both layers)
    float* W1L  = F    + (size_t)BDIM * F1K;   // 2304*256
    float* out1 = W1L  + (size_t)F1K * HID;    // 4096*256
    float* W2L  = out1 + (size_t)BDIM * HID;   // 2304*128

    // 1) weight preprocessing
    transpose_w1<<<(INDIM * HID + 255) / 256, 256, 0, stream>>>(W1, W1T);
    pack_kan_w<<<(HID * HID + 255) / 256, 256, 0, stream>>>(coef1, sb1, sp1, mask1, W1L, HID, HID);
    pack_kan_w<<<(HID * OUTDIM + 255) / 256, 256, 0, stream>>>(coef2, sb2, sp2, mask2, W2L, HID, OUTDIM);

    // 2) h = x @ W1^T + b1      (4096x1024 * 1024x256)
    gemm_f32_wmma<<<dim3(HID / BLK_N, BDIM / BLK_M), 256, 0, stream>>>(x, W1T, b1, h, BDIM, HID, INDIM);

    // 3) layer 1: featurize + GEMM
    kan_featurize<<<(BDIM * HID + 255) / 256, 256, 0, stream>>>(h, grid1, F, BDIM, HID);
    gemm_f32_wmma<<<dim3(HID / BLK_N, BDIM / BLK_M), 256, 0, stream>>>(F, W1L, nullptr, out1, BDIM, HID, F1K);

    // 4) layer 2: featurize + GEMM -> d_out
    kan_featurize<<<(BDIM * HID + 255) / 256, 256, 0, stream>>>(out1, grid2, F, BDIM, HID);
    gemm_f32_wmma<<<dim3(OUTDIM / BLK_N, BDIM / BLK_M), 256, 0, stream>>>(F, W2L, nullptr, out, BDIM, OUTDIM, F1K);
}